// MaskedDenseLayer_10582799417527
// MI455X (gfx1250) — compile-verified
//
#include <hip/hip_runtime.h>

#define BS      512
#define IN_DIM  784
#define OUT_DIM 1024
#define N_MASKS 16

#define NT   64    // output columns staged per workgroup
#define MT   16    // batch rows per WMMA M-tile
#define LSTR 788   // LDS row stride in floats: mult of 4 (16B-aligned b128), 788%64=20 spreads banks

typedef __attribute__((ext_vector_type(2))) float v2f;
typedef __attribute__((ext_vector_type(8))) float v8f;

// ---------------------------------------------------------------------------
// Kernel 1: bucket batch rows by their state id (16 buckets).
// Slot order in a bucket is nondeterministic, but each row's result is
// computed identically regardless of slot -> output is deterministic.
// ---------------------------------------------------------------------------
__global__ void made_gather(const int* __restrict__ state,
                            int* __restrict__ counts,
                            int* __restrict__ lists) {
  int b = blockIdx.x * blockDim.x + threadIdx.x;
  if (b < BS) {
    int s = state[b];
    int slot = atomicAdd(&counts[s], 1);
    lists[s * BS + slot] = b;
  }
}

// ---------------------------------------------------------------------------
// Kernel 2: one workgroup per (state, 64-wide output strip).
//  - Stage MK = kernel .* masks[s] strip (784 x 64 f32, ~202KB) in LDS once:
//    masks stream from HBM exactly once chip-wide (the 51.4MB roofline term).
//  - 256 threads = 8 waves: waves 0-3 compute M-tile mb (4 x 16-col subtiles),
//    waves 4-7 compute M-tile mb+1. Two x-tiles live in LDS (~101KB).
//  - Per wave: K=784 via v_wmma_f32_16x16x4_f32 with 2 accumulators to break
//    the WMMA RAW chain (2 waves/SIMD further hide latency).
// ---------------------------------------------------------------------------
__global__ __launch_bounds__(256, 1)
void made_masked_gemm(const float* __restrict__ x,
                      const float* __restrict__ masks,
                      const float* __restrict__ kmat,
                      const float* __restrict__ b0,
                      const int* __restrict__ counts,
                      const int* __restrict__ lists,
                      float* __restrict__ out) {
  __shared__ float lds_mk[NT * LSTR];      // masked kernel, [o][k] rows
  __shared__ float lds_xt[2 * MT * LSTR];  // two 16-row x tiles, [r][k]

  const int s  = blockIdx.y;
  const int ob = blockIdx.x * NT;
  const int count = counts[s];
  if (count == 0) return;

  const int t     = threadIdx.x;
  const int lane  = t & 31;
  const int wave  = t >> 5;            // 0..7
  const int nsub  = wave & 3;          // 16-col subtile
  const int msel  = wave >> 2;         // which of the two staged M-tiles
  const int laneN = lane & 15;
  const int koff  = (lane >> 4) * 2;   // A/B frag: low lanes K+{0,1}, high K+{2,3}

  // ---- Stage masked-kernel strip: 4 k's per thread -> one b128 LDS store ----
  for (int u = t; u < NT * (IN_DIM / 4); u += 256) {
    int o  = u & (NT - 1);             // consecutive threads -> coalesced global
    int k  = (u >> 6) * 4;
    size_t g = (size_t)k * OUT_DIM + (size_t)(ob + o);
    size_t gm = (size_t)s * IN_DIM * OUT_DIM + g;
    float4 v;
    v.x = kmat[g]              * masks[gm];
    v.y = kmat[g + OUT_DIM]    * masks[gm + OUT_DIM];
    v.z = kmat[g + 2*OUT_DIM]  * masks[gm + 2*OUT_DIM];
    v.w = kmat[g + 3*OUT_DIM]  * masks[gm + 3*OUT_DIM];
    *(float4*)&lds_mk[o * LSTR + k] = v;   // (o*788+k)*4 is 16B aligned
  }

  const int* __restrict__ list = lists + s * BS;
  const int numM = (count + MT - 1) / MT;
  const int col  = ob + 16 * nsub + laneN;
  const float bias = b0[col];

  for (int mb = 0; mb < numM; mb += 2) {
    // ---- Stage up to 32 x-rows (two tiles, zero-padded) as [r][k] ----
    for (int u = t; u < 2 * MT * (IN_DIM / 4); u += 256) {
      int r = u / (IN_DIM / 4);
      int k = (u % (IN_DIM / 4)) * 4;
      int slot = mb * MT + r;
      float4 v = {0.f, 0.f, 0.f, 0.f};
      if (slot < count) {
        int grow = list[slot];
        v = *(const float4*)(x + (size_t)grow * IN_DIM + k);  // 16B aligned
      }
      *(float4*)&lds_xt[r * LSTR + k] = v;
    }
    __syncthreads();   // MK (first pass) + both XT tiles visible

    const int m = mb + msel;
    if (m < numM) {
      // ---- 16x16 tile: 196 f32 WMMAs, 2 accumulators (RAW distance 2) ----
      v8f acc0 = {}, acc1 = {};
      const float* __restrict__ arow =
          &lds_xt[(msel * MT + laneN) * LSTR + koff];
      const float* __restrict__ brow =
          &lds_mk[(16 * nsub + laneN) * LSTR + koff];
      for (int k = 0; k < IN_DIM; k += 8) {
        v2f a0, b0f, a1, b1f;
        a0.x  = arow[k];     a0.y  = arow[k + 1];
        b0f.x = brow[k];     b0f.y = brow[k + 1];
        a1.x  = arow[k + 4]; a1.y  = arow[k + 5];
        b1f.x = brow[k + 4]; b1f.y = brow[k + 5];
        acc0 = __builtin_amdgcn_wmma_f32_16x16x4_f32(
            false, a0, false, b0f, (short)0, acc0, false, false);
        acc1 = __builtin_amdgcn_wmma_f32_16x16x4_f32(
            false, a1, false, b1f, (short)0, acc1, false, false);
      }
      v8f acc = acc0 + acc1;

      // ---- Epilogue: C/D layout VGPR rr = row (rr + 8*laneHi), col = laneN.
      //      Unconditional clamped list load, predicated store only. ----
      const int base = m * MT + ((lane >> 4) * 8);
      for (int rr = 0; rr < 8; ++rr) {
        int slot = base + rr;
        int cslot = slot < count - 1 ? slot : count - 1;
        int grow = list[cslot];
        float v = fmaxf(acc[rr] + bias, 0.0f);
        if (slot < count) out[(size_t)grow * OUT_DIM + col] = v;
      }
    }
    __syncthreads();   // all waves done reading lds_xt before restaging
  }
}

// ---------------------------------------------------------------------------
extern "C" void kernel_launch(void* const* d_in, const int* in_sizes, int n_in,
                              void* d_out, int out_size, void* d_ws, size_t ws_size,
                              hipStream_t stream) {
  const float* x     = (const float*)d_in[0];  // [512,784]
  const int*   state = (const int*)  d_in[1];  // [512,1]
  const float* masks = (const float*)d_in[2];  // [16,784,1024]
  const float* kmat  = (const float*)d_in[3];  // [784,1024]
  const float* b0    = (const float*)d_in[4];  // [1,1024]
  float* out = (float*)d_out;                  // [512,1024]

  int* counts = (int*)d_ws;                    // 16 ints
  int* lists  = counts + 16;                   // 16*512 ints

  hipMemsetAsync(counts, 0, N_MASKS * sizeof(int), stream);
  made_gather<<<dim3((BS + 127) / 128), dim3(128), 0, stream>>>(state, counts, lists);
  made_masked_gemm<<<dim3(OUT_DIM / NT, N_MASKS), dim3(256), 0, stream>>>(
      x, masks, kmat, b0, counts, lists, out);
}